// GraphConvNet_2104533975239
// MI455X (gfx1250) — compile-verified
//
#include <hip/hip_runtime.h>
#include <hip/hip_bf16.h>
#include <math.h>

#define N_NODES 100000
#define N_EDGES 1600000
#define N_GRAPHS 64
#define D 128

typedef float v2f __attribute__((ext_vector_type(2)));
typedef float v8f __attribute__((ext_vector_type(8)));

// ---------------- utility fill ----------------
__global__ void fill_zero_f32(float* __restrict__ p, int n) {
    int i = blockIdx.x * blockDim.x + threadIdx.x;
    if (i < n) p[i] = 0.0f;
}

// ---------------- degree accumulation ----------------
__global__ void degree_kernel(const int* __restrict__ src, const int* __restrict__ dst,
                              float* __restrict__ deg_out, float* __restrict__ deg_in, int E) {
    int e = blockIdx.x * blockDim.x + threadIdx.x;
    if (e >= E) return;
    atomicAdd(&deg_out[src[e]], 1.0f);
    atomicAdd(&deg_in[dst[e]], 1.0f);
}

__global__ void norm_kernel(const float* __restrict__ deg_out, const float* __restrict__ deg_in,
                            float* __restrict__ norm_src, float* __restrict__ norm_dst, int n) {
    int i = blockIdx.x * blockDim.x + threadIdx.x;
    if (i >= n) return;
    float o = deg_out[i]; o = o < 1.0f ? 1.0f : o;
    float d = deg_in[i];  d = d < 1.0f ? 1.0f : d;
    norm_src[i] = rsqrtf(o);
    norm_dst[i] = rsqrtf(d);
}

// ---------------- edge scatter: agg[dst] += h[src] * norm_src[src] ----------------
// One thread handles one edge x 4 consecutive features (float4 load, 4 atomics).
__global__ void scatter_kernel(const float* __restrict__ h, const float* __restrict__ norm_src,
                               const int* __restrict__ src, const int* __restrict__ dst,
                               float* __restrict__ agg, int E) {
    int idx = blockIdx.x * blockDim.x + threadIdx.x;   // E * (D/4) threads
    int e = idx >> 5;                                  // D/4 == 32 chunks per edge
    if (e >= E) return;
    int q = (idx & 31) << 2;                           // feature base (0,4,...,124)
    int s = src[e];
    int d = dst[e];
    float scale = norm_src[s];
    const float4 x = *reinterpret_cast<const float4*>(h + (size_t)s * D + q);
    float* ap = agg + (size_t)d * D + q;
    atomicAdd(ap + 0, x.x * scale);
    atomicAdd(ap + 1, x.y * scale);
    atomicAdd(ap + 2, x.z * scale);
    atomicAdd(ap + 3, x.w * scale);
}

// ---------------- WMMA GEMM: out = (agg * norm_dst) @ W + b ----------------
// Block: 128 threads (4 waves). Block tile: 16 nodes x 128 output cols.
// Each wave computes two 16x16 output tiles via 32x v_wmma_f32_16x16x4_f32 (K=128).
// LDS row padded to 132 floats so A-fragment reads (M-strided) hit 16 distinct banks.
__global__ __launch_bounds__(128) void gemm_wmma_kernel(
    const float* __restrict__ agg,      // [N, 128]
    const float* __restrict__ norm_dst, // [N]
    const float* __restrict__ W,        // [128, 128] row-major (in, out)
    const float* __restrict__ bias,     // [128]
    float* __restrict__ out)            // [N, 128]
{
    __shared__ float Xs[16][D + 4];

    const int m0 = blockIdx.x * 16;
    const int t  = threadIdx.x;          // 0..127

    // Stage X tile (scaled by norm_dst) into LDS; column t, all 16 rows (coalesced).
    #pragma unroll
    for (int r = 0; r < 16; ++r) {
        int node = m0 + r;
        Xs[r][t] = agg[(size_t)node * D + t] * norm_dst[node];
    }
    __syncthreads();

    const int wave  = t >> 5;            // 0..3
    const int lane  = t & 31;
    const int mrow  = lane & 15;         // A-matrix row for this lane
    const int khalf = (lane >> 4) << 1;  // 0 (lanes 0-15) or 2 (lanes 16-31)

    #pragma unroll
    for (int sub = 0; sub < 2; ++sub) {
        const int nt   = wave * 2 + sub;     // 0..7
        const int n0   = nt * 16;
        const int ncol = n0 + (lane & 15);   // B/D column for this lane

        v8f c = {};
        #pragma unroll
        for (int k = 0; k < D; k += 4) {
            const int kk = k + khalf;
            v2f a, b;
            // A 16x4 fragment: lane holds (M=mrow, K=kk) and (M=mrow, K=kk+1)
            a.x = Xs[mrow][kk];
            a.y = Xs[mrow][kk + 1];
            // B 4x16 fragment: lane holds (K=kk, N=ncol) and (K=kk+1, N=ncol)
            b.x = W[(size_t)kk * D + ncol];
            b.y = W[(size_t)(kk + 1) * D + ncol];
            c = __builtin_amdgcn_wmma_f32_16x16x4_f32(
                    /*neg_a=*/false, a, /*neg_b=*/false, b,
                    /*c_mod=*/(short)0, c, /*reuse_a=*/false, /*reuse_b=*/false);
        }

        // D layout: VGPR v -> row m0 + v + 8*(lane>=16), col ncol.
        const int rbase = m0 + ((lane >> 4) << 3);
        const float bb = bias[ncol];
        #pragma unroll
        for (int v = 0; v < 8; ++v) {
            out[(size_t)(rbase + v) * D + ncol] = c[v] + bb;
        }
    }
}

// ---------------- per-graph pooling ----------------
__global__ void pool_kernel(const float* __restrict__ h, const int* __restrict__ gid,
                            float* __restrict__ sums, float* __restrict__ cnts, int n_nodes) {
    int idx = blockIdx.x * blockDim.x + threadIdx.x;   // n_nodes * D
    int n = idx >> 7;
    if (n >= n_nodes) return;
    int d = idx & (D - 1);
    int g = gid[n];
    atomicAdd(&sums[(size_t)g * D + d], h[idx]);
    if (d == 0) atomicAdd(&cnts[g], 1.0f);
}

// ---------------- final FC + sigmoid ----------------
__global__ __launch_bounds__(128) void head_kernel(
    const float* __restrict__ sums, const float* __restrict__ cnts,
    const float* __restrict__ fc_w, const float* __restrict__ fc_b,
    float* __restrict__ out)
{
    __shared__ float red[D];
    const int g = blockIdx.x;
    const int d = threadIdx.x;
    float cnt = cnts[g];
    cnt = cnt < 1.0f ? 1.0f : cnt;
    red[d] = (sums[(size_t)g * D + d] / cnt) * fc_w[d];
    __syncthreads();
    #pragma unroll
    for (int s = D / 2; s > 0; s >>= 1) {
        if (d < s) red[d] += red[d + s];
        __syncthreads();
    }
    if (d == 0) {
        float z = red[0] + fc_b[0];
        out[g] = 1.0f / (1.0f + __expf(-z));
    }
}

extern "C" void kernel_launch(void* const* d_in, const int* in_sizes, int n_in,
                              void* d_out, int out_size, void* d_ws, size_t ws_size,
                              hipStream_t stream) {
    const float* h0   = (const float*)d_in[0];
    const int*   src  = (const int*)d_in[1];
    const int*   dst  = (const int*)d_in[2];
    const int*   gid  = (const int*)d_in[3];
    const float* Ws[5] = { (const float*)d_in[4],  (const float*)d_in[6],
                           (const float*)d_in[8],  (const float*)d_in[10],
                           (const float*)d_in[12] };
    const float* Bs[5] = { (const float*)d_in[5],  (const float*)d_in[7],
                           (const float*)d_in[9],  (const float*)d_in[11],
                           (const float*)d_in[13] };
    const float* fcw = (const float*)d_in[14];
    const float* fcb = (const float*)d_in[15];
    float* out = (float*)d_out;

    // Workspace layout
    char* ws = (char*)d_ws;
    float* bufH   = (float*)ws;  ws += (size_t)N_NODES * D * sizeof(float); // layer outputs
    float* bufAgg = (float*)ws;  ws += (size_t)N_NODES * D * sizeof(float); // aggregation
    float* degO   = (float*)ws;  ws += (size_t)N_NODES * sizeof(float);
    float* degI   = (float*)ws;  ws += (size_t)N_NODES * sizeof(float);
    float* nrmS   = (float*)ws;  ws += (size_t)N_NODES * sizeof(float);
    float* nrmD   = (float*)ws;  ws += (size_t)N_NODES * sizeof(float);
    float* sums   = (float*)ws;  ws += (size_t)N_GRAPHS * D * sizeof(float);
    float* cnts   = (float*)ws;  ws += (size_t)N_GRAPHS * sizeof(float);

    const int TB = 256;

    // Degrees + norms (degO/degI contiguous -> one zero fill over 2N)
    fill_zero_f32<<<(2 * N_NODES + TB - 1) / TB, TB, 0, stream>>>(degO, 2 * N_NODES);
    degree_kernel<<<(N_EDGES + TB - 1) / TB, TB, 0, stream>>>(src, dst, degO, degI, N_EDGES);
    norm_kernel<<<(N_NODES + TB - 1) / TB, TB, 0, stream>>>(degO, degI, nrmS, nrmD, N_NODES);

    const int nd_elems = N_NODES * D;                       // 12.8M
    const int scatter_threads = N_EDGES * (D / 4) / 4 * 4;  // E * 32
    const float* hcur = h0;
    for (int l = 0; l < 5; ++l) {
        fill_zero_f32<<<(nd_elems + TB - 1) / TB, TB, 0, stream>>>(bufAgg, nd_elems);
        scatter_kernel<<<(scatter_threads + TB - 1) / TB, TB, 0, stream>>>(
            hcur, nrmS, src, dst, bufAgg, N_EDGES);
        gemm_wmma_kernel<<<N_NODES / 16, 128, 0, stream>>>(
            bufAgg, nrmD, Ws[l], Bs[l], bufH);
        hcur = bufH;
    }

    // Pooling + head (sums and cnts contiguous -> one zero fill)
    fill_zero_f32<<<(N_GRAPHS * D + N_GRAPHS + TB - 1) / TB, TB, 0, stream>>>(
        sums, N_GRAPHS * D + N_GRAPHS);
    pool_kernel<<<(nd_elems + TB - 1) / TB, TB, 0, stream>>>(bufH, gid, sums, cnts, N_NODES);
    head_kernel<<<N_GRAPHS, D, 0, stream>>>(sums, cnts, fcw, fcb, out);
}